// GAT_71880572666191
// MI455X (gfx1250) — compile-verified
//
#include <hip/hip_runtime.h>

typedef float v2f __attribute__((ext_vector_type(2)));
typedef float v4f __attribute__((ext_vector_type(4)));
typedef float v8f __attribute__((ext_vector_type(8)));

#define OUTD 64

// ---------------- degree / normalization ----------------

__global__ void gcn_init_deg(float* __restrict__ deg, int n) {
  int i = blockIdx.x * blockDim.x + threadIdx.x;
  if (i < n) deg[i] = 1.0f;  // self-loop
}

__global__ void gcn_deg_accum(const long long* __restrict__ ei, float* __restrict__ deg, int nE) {
  int e = blockIdx.x * blockDim.x + threadIdx.x;
  if (e < nE) {
    int d = (int)ei[(size_t)nE + e];  // dst row
    atomicAdd(&deg[d], 1.0f);
  }
}

__global__ void gcn_dinv(float* __restrict__ deg, int n) {
  int i = blockIdx.x * blockDim.x + threadIdx.x;
  if (i < n) deg[i] = rsqrtf(deg[i]);
}

// ---------------- WMMA fp32 GEMM with fused self-loop epilogue ----------------
// C[nRows,64]    = (RELU_A ? relu(A) : A)[nRows,K] @ B[K,64]
// Cagg[nRows,64] = C * dinv[row]^2          (self-loop term; initializes scatter buffer)
//
// One wave -> one 16x16 output tile; K-loop of v_wmma_f32_16x16x4_f32.
// A frag (16x4 f32): lanes 0-15 hold row M=lane, K={0,1}; lanes 16-31 K={2,3}.
// B frag (4x16 f32): lanes 0-15 hold col N=lane, K={0,1}; lanes 16-31 K={2,3}.
// C/D  (16x16 f32): VGPR r, lanes 0-15 -> M=r, N=lane; lanes 16-31 -> M=r+8.

template <int KDIM, bool RELU_A>
__global__ void gcn_gemm_wmma(const float* __restrict__ A, const float* __restrict__ B,
                              const float* __restrict__ dinv, float* __restrict__ C,
                              float* __restrict__ Cagg, int nRows) {
  const int lane = threadIdx.x & 31;
  const int wave = threadIdx.x >> 5;
  const int tile = blockIdx.x * 8 + wave;          // 8 waves / 256-thread block
  const int nRowTiles = (nRows + 15) >> 4;
  if (tile >= nRowTiles * 4) return;               // wave-uniform guard (EXEC stays full)

  const int m0 = (tile >> 2) << 4;                 // row tile * 16
  const int n0 = (tile & 3) << 4;                  // col tile * 16 (64/16 = 4 tiles)
  const int half = lane >> 4;                      // 0: K={0,1}, 1: K={2,3}
  const int idx = lane & 15;                       // row (A) / col (B,C) within tile
  const int kk = half << 1;

  int arow = m0 + idx;
  if (arow >= nRows) arow = nRows - 1;             // clamp (no EXEC divergence)
  const float* __restrict__ Arow = A + (size_t)arow * KDIM;
  const float* __restrict__ Bcol = B + n0 + idx;

  v8f acc = {};
#pragma unroll
  for (int k = 0; k < KDIM; k += 4) {
    v2f a = *(const v2f*)(Arow + k + kk);          // 8B-aligned: k%4==0, kk even
    if (RELU_A) {
      a.x = fmaxf(a.x, 0.0f);
      a.y = fmaxf(a.y, 0.0f);
    }
    v2f b;
    b.x = Bcol[(size_t)(k + kk) * OUTD];
    b.y = Bcol[(size_t)(k + kk + 1) * OUTD];
    acc = __builtin_amdgcn_wmma_f32_16x16x4_f32(
        /*neg_a=*/false, a, /*neg_b=*/false, b,
        /*c_mod=*/(short)0, acc, /*reuse_a=*/false, /*reuse_b=*/false);
  }

  // fused epilogue: write h and the self-loop-scaled aggregation seed
#pragma unroll
  for (int r = 0; r < 8; ++r) {
    int row = m0 + r + (half << 3);
    if (row < nRows) {
      float v = acc[r];
      size_t off = (size_t)row * OUTD + n0 + idx;
      C[off] = v;
      float w = dinv[row];                         // uniform per 16-lane group -> broadcast
      Cagg[off] = v * (w * w);
    }
  }
}

// ---------------- edge scatter: out[dst,:] += h[src,:] * dinv[src]*dinv[dst] ----------------
// 16 threads per edge, float4 gather + 4x global_atomic_add_f32 scatter.

__global__ void gcn_scatter(const long long* __restrict__ ei, const float* __restrict__ h,
                            const float* __restrict__ dinv, float* __restrict__ out, int nE) {
  int gid = blockIdx.x * blockDim.x + threadIdx.x;
  int e = gid >> 4;
  if (e >= nE) return;
  int f = (gid & 15) << 2;
  int s = (int)ei[e];
  int d = (int)ei[(size_t)nE + e];
  float w = dinv[s] * dinv[d];
  v4f hv = *(const v4f*)(h + (size_t)s * OUTD + f);
  float* o = out + (size_t)d * OUTD + f;
  atomicAdd(o + 0, hv.x * w);
  atomicAdd(o + 1, hv.y * w);
  atomicAdd(o + 2, hv.z * w);
  atomicAdd(o + 3, hv.w * w);
}

// ---------------- launch ----------------

extern "C" void kernel_launch(void* const* d_in, const int* in_sizes, int n_in,
                              void* d_out, int out_size, void* d_ws, size_t ws_size,
                              hipStream_t stream) {
  const float* x = (const float*)d_in[0];
  const long long* ei = (const long long*)d_in[1];  // [2, E] int64
  const float* W1 = (const float*)d_in[2];          // [128, 64]
  const float* W2 = (const float*)d_in[3];          // [64, 64]
  float* out = (float*)d_out;                       // [N, 64]

  const int N = in_sizes[0] / 128;
  const int E = in_sizes[1] / 2;

  // workspace layout (floats): dinv[N] | h1[N*64] | agg1[N*64] | h3[N*64]
  float* ws = (float*)d_ws;
  size_t nAl = ((size_t)N + 255) & ~(size_t)255;
  float* dinv = ws;
  float* h1 = ws + nAl;
  float* agg1 = h1 + nAl * OUTD;
  float* h3 = agg1 + nAl * OUTD;

  const int T = 256;
  // degrees + normalization (deg computed in-place, then dinv = rsqrt(deg))
  gcn_init_deg<<<(N + T - 1) / T, T, 0, stream>>>(dinv, N);
  gcn_deg_accum<<<(E + T - 1) / T, T, 0, stream>>>(ei, dinv, E);
  gcn_dinv<<<(N + T - 1) / T, T, 0, stream>>>(dinv, N);

  const int nRowTiles = (N + 15) / 16;
  const int gemmBlocks = (nRowTiles * 4 + 7) / 8;   // 8 waves per block, 4 col-tiles
  const int edgeBlocks = (int)(((size_t)E * 16 + T - 1) / T);

  // layer 1: h1 = x @ W1 ; agg1 = dinv^2*h1 (fused) + scatter of neighbor messages
  gcn_gemm_wmma<128, false><<<gemmBlocks, T, 0, stream>>>(x, W1, dinv, h1, agg1, N);
  gcn_scatter<<<edgeBlocks, T, 0, stream>>>(ei, h1, dinv, agg1, E);

  // layer 2: h3 = relu(agg1) @ W2 ; out = dinv^2*h3 (fused) + scatter
  gcn_gemm_wmma<64, true><<<gemmBlocks, T, 0, stream>>>(agg1, W2, dinv, h3, out, N);
  gcn_scatter<<<edgeBlocks, T, 0, stream>>>(ei, h3, dinv, out, E);
}